// CrossAttentionConditioner_45208825757708
// MI455X (gfx1250) — compile-verified
//
#include <hip/hip_runtime.h>
#include <math.h>

#define N_EDGES   4096
#define HIDDEN    256
#define EDGE_DIM  128
#define NUM_HEADS 4
#define HEAD_DIM  64
#define NUM_GRAPHS 64

typedef __attribute__((ext_vector_type(2))) float v2f;
typedef __attribute__((ext_vector_type(8))) float v8f;

// ---------------------------------------------------------------------------
// Fuse edge projection into K/V in-projections:
//   Wk_eff = wk @ w_edge   [256,128] ;  bk_eff = wk @ b_edge + bk
//   Wv_eff = wv @ w_edge   [256,128] ;  bv_eff = wv @ b_edge + bv
// w_in is [768,256] packed (wq rows 0..255, wk 256..511, wv 512..767).
// ---------------------------------------------------------------------------
__global__ void fuse_weights_kernel(const float* __restrict__ w_in,
                                    const float* __restrict__ b_in,
                                    const float* __restrict__ w_edge,
                                    const float* __restrict__ b_edge,
                                    float* __restrict__ wk_eff,
                                    float* __restrict__ wv_eff,
                                    float* __restrict__ bk_eff,
                                    float* __restrict__ bv_eff) {
    const int WELEMS = HIDDEN * EDGE_DIM;           // 32768
    int t = blockIdx.x * blockDim.x + threadIdx.x;
    if (t < 2 * WELEMS) {
        int which = t / WELEMS;                     // 0 = K, 1 = V
        int idx   = t % WELEMS;
        int n = idx / EDGE_DIM;
        int k = idx % EDGE_DIM;
        const float* w = w_in + (HIDDEN + which * HIDDEN + n) * HIDDEN;
        float s = 0.f;
        #pragma unroll 8
        for (int j = 0; j < HIDDEN; ++j)
            s = fmaf(w[j], w_edge[j * EDGE_DIM + k], s);
        (which ? wv_eff : wk_eff)[idx] = s;
    } else if (t < 2 * WELEMS + 2 * HIDDEN) {
        int u = t - 2 * WELEMS;
        int which = u / HIDDEN;
        int n     = u % HIDDEN;
        const float* w = w_in + (HIDDEN + which * HIDDEN + n) * HIDDEN;
        float s = b_in[HIDDEN + which * HIDDEN + n];
        #pragma unroll 8
        for (int j = 0; j < HIDDEN; ++j)
            s = fmaf(w[j], b_edge[j], s);
        (which ? bv_eff : bk_eff)[n] = s;
    }
}

// ---------------------------------------------------------------------------
// Segment boundaries: seg_start[g] = lower_bound(gidx, g); seg_start[64] = N.
// ---------------------------------------------------------------------------
__global__ void seg_starts_kernel(const int* __restrict__ gidx,
                                  int* __restrict__ seg_start) {
    int g = blockIdx.x * blockDim.x + threadIdx.x;
    if (g > NUM_GRAPHS) return;
    if (g == NUM_GRAPHS) { seg_start[g] = N_EDGES; return; }
    int lo = 0, hi = N_EDGES;
    while (lo < hi) {
        int mid = (lo + hi) >> 1;
        if (gidx[mid] < g) lo = mid + 1; else hi = mid;
    }
    seg_start[g] = lo;
}

// ---------------------------------------------------------------------------
// C[M,N] = A[M,K] @ W[N,K]^T + bias[N] (+ resid[M,N])
// One wave per 16x16 output tile using V_WMMA_F32_16X16X4_F32.
//
// Per-lane operand layout (ISA 7.12.2, 32-bit data, wave32):
//   half = lane>>4, l16 = lane&15
//   A (16x4, v2f):  a[j]  = A[tile_m + l16][kb + j + 2*half]       (j = 0,1)
//   B (4x16, v2f):  b[j]  = W[tile_n + l16][kb + j + 2*half]       (Bt[k][n]=W[n][k])
//   D (16x16, v8f): d[r] -> C[tile_m + r + 8*half][tile_n + l16]
// ---------------------------------------------------------------------------
__global__ void gemm_wmma_f32_kernel(const float* __restrict__ A,
                                     const float* __restrict__ W,
                                     const float* __restrict__ bias,
                                     const float* __restrict__ resid,
                                     float* __restrict__ C,
                                     int M, int N, int K) {
    int wave = (blockIdx.x * blockDim.x + threadIdx.x) >> 5;
    int lane = threadIdx.x & 31;
    int ntiles = N >> 4;
    int mtiles = M >> 4;
    if (wave >= mtiles * ntiles) return;       // uniform per wave
    int tm = (wave / ntiles) << 4;
    int tn = (wave % ntiles) << 4;

    int half = lane >> 4;
    int l16  = lane & 15;

    const float* arow = A + (size_t)(tm + l16) * K + 2 * half;
    const float* wrow = W + (size_t)(tn + l16) * K + 2 * half;

    v8f acc = {0.f, 0.f, 0.f, 0.f, 0.f, 0.f, 0.f, 0.f};
    for (int k = 0; k < K; k += 4) {
        v2f a = *(const v2f*)(arow + k);
        v2f b = *(const v2f*)(wrow + k);
        acc = __builtin_amdgcn_wmma_f32_16x16x4_f32(
            /*neg_a=*/false, a, /*neg_b=*/false, b,
            /*c_mod=*/(short)0, acc, /*reuse_a=*/false, /*reuse_b=*/false);
    }

    int col = tn + l16;
    float bval = bias ? bias[col] : 0.f;
    #pragma unroll
    for (int r = 0; r < 8; ++r) {
        int row = tm + r + 8 * half;
        float v = acc[r] + bval;
        size_t off = (size_t)row * N + col;
        if (resid) v += resid[off];
        C[off] = v;
    }
}

// ---------------------------------------------------------------------------
// Block-diagonal attention, one thread per (row, head), online softmax.
// Q/K/V buffers are [N, HIDDEN]; head h occupies columns [h*64, h*64+64).
// Exact per-segment softmax (mathematically equal to the masked dense one).
// ---------------------------------------------------------------------------
__global__ void attention_kernel(const float* __restrict__ Q,
                                 const float* __restrict__ Kb,
                                 const float* __restrict__ Vb,
                                 const int* __restrict__ gidx,
                                 const int* __restrict__ seg_start,
                                 float* __restrict__ O) {
    int t = blockIdx.x * blockDim.x + threadIdx.x;
    if (t >= N_EDGES * NUM_HEADS) return;
    int h = t / N_EDGES;
    int n = t % N_EDGES;

    int g  = gidx[n];
    int s0 = seg_start[g];
    int s1 = seg_start[g + 1];

    const float scale = 0.125f;                   // 1/sqrt(64)
    float q[HEAD_DIM];
    const float* qp = Q + (size_t)n * HIDDEN + h * HEAD_DIM;
    #pragma unroll
    for (int d = 0; d < HEAD_DIM; ++d) q[d] = qp[d] * scale;

    float acc[HEAD_DIM];
    #pragma unroll
    for (int d = 0; d < HEAD_DIM; ++d) acc[d] = 0.f;

    float mmax  = -INFINITY;
    float denom = 0.f;

    for (int m = s0; m < s1; ++m) {
        const float* kp = Kb + (size_t)m * HIDDEN + h * HEAD_DIM;
        float s = 0.f;
        #pragma unroll
        for (int d = 0; d < HEAD_DIM; ++d) s = fmaf(q[d], kp[d], s);

        float mnew = fmaxf(mmax, s);
        float corr = __expf(mmax - mnew);         // exp(-inf)=0 handles first iter
        float p    = __expf(s - mnew);
        denom = denom * corr + p;

        const float* vp = Vb + (size_t)m * HIDDEN + h * HEAD_DIM;
        #pragma unroll
        for (int d = 0; d < HEAD_DIM; ++d) acc[d] = fmaf(acc[d], corr, p * vp[d]);
        mmax = mnew;
    }

    float inv = 1.f / denom;
    float* op = O + (size_t)n * HIDDEN + h * HEAD_DIM;
    #pragma unroll
    for (int d = 0; d < HEAD_DIM; ++d) op[d] = acc[d] * inv;
}

// ---------------------------------------------------------------------------
extern "C" void kernel_launch(void* const* d_in, const int* in_sizes, int n_in,
                              void* d_out, int out_size, void* d_ws, size_t ws_size,
                              hipStream_t stream) {
    const float* query  = (const float*)d_in[0];   // [4096,256]
    const float* key    = (const float*)d_in[1];   // [4096,128]
    const float* value  = (const float*)d_in[2];   // [4096,128]
    const int*   gidx   = (const int*)  d_in[3];   // [4096]
    const float* w_edge = (const float*)d_in[4];   // [256,128]
    const float* b_edge = (const float*)d_in[5];   // [256]
    const float* w_in   = (const float*)d_in[6];   // [768,256]
    const float* b_in   = (const float*)d_in[7];   // [768]
    const float* w_out  = (const float*)d_in[8];   // [256,256]
    const float* b_out  = (const float*)d_in[9];   // [256]
    float* out = (float*)d_out;                    // [4096,256]

    const size_t NH = (size_t)N_EDGES * HIDDEN;    // 1048576 floats
    float* ws     = (float*)d_ws;
    float* Qbuf   = ws;
    float* Kbuf   = Qbuf + NH;
    float* Vbuf   = Kbuf + NH;
    float* Abuf   = Vbuf + NH;
    float* wk_eff = Abuf + NH;
    float* wv_eff = wk_eff + HIDDEN * EDGE_DIM;
    float* bk_eff = wv_eff + HIDDEN * EDGE_DIM;
    float* bv_eff = bk_eff + HIDDEN;
    int*   seg    = (int*)(bv_eff + HIDDEN);

    // 1) fuse edge projection into K/V in-projection weights
    {
        int total = 2 * HIDDEN * EDGE_DIM + 2 * HIDDEN;
        int blocks = (total + 127) / 128;
        fuse_weights_kernel<<<blocks, 128, 0, stream>>>(
            w_in, b_in, w_edge, b_edge, wk_eff, wv_eff, bk_eff, bv_eff);
    }

    // 2) segment boundaries
    seg_starts_kernel<<<1, 128, 0, stream>>>(gidx, seg);

    // 3) Q = query @ wq^T + bq   (M=4096, N=256, K=256) -> 4096 waves
    {
        int waves = (N_EDGES / 16) * (HIDDEN / 16);
        gemm_wmma_f32_kernel<<<waves * 32 / 256, 256, 0, stream>>>(
            query, w_in, b_in, nullptr, Qbuf, N_EDGES, HIDDEN, HIDDEN);
    }
    // 4) K = key @ Wk_eff^T + bk_eff   (K=128)
    {
        int waves = (N_EDGES / 16) * (HIDDEN / 16);
        gemm_wmma_f32_kernel<<<waves * 32 / 256, 256, 0, stream>>>(
            key, wk_eff, bk_eff, nullptr, Kbuf, N_EDGES, HIDDEN, EDGE_DIM);
    }
    // 5) V = value @ Wv_eff^T + bv_eff
    {
        int waves = (N_EDGES / 16) * (HIDDEN / 16);
        gemm_wmma_f32_kernel<<<waves * 32 / 256, 256, 0, stream>>>(
            value, wv_eff, bv_eff, nullptr, Vbuf, N_EDGES, HIDDEN, EDGE_DIM);
    }

    // 6) per-segment attention
    {
        int total = N_EDGES * NUM_HEADS;
        attention_kernel<<<(total + 127) / 128, 128, 0, stream>>>(
            Qbuf, Kbuf, Vbuf, gidx, seg, Abuf);
    }

    // 7) out = Abuf @ w_out^T + b_out + query
    {
        int waves = (N_EDGES / 16) * (HIDDEN / 16);
        gemm_wmma_f32_kernel<<<waves * 32 / 256, 256, 0, stream>>>(
            Abuf, w_out, b_out, query, out, N_EDGES, HIDDEN, HIDDEN);
    }
}